// TripletLoss_biu2_47012712022533
// MI455X (gfx1250) — compile-verified
//
#include <hip/hip_runtime.h>
#include <hip/hip_bf16.h>
#include <math.h>

// Problem constants (match reference)
constexpr int N = 4096;
constexpr int D = 4096;
constexpr float F_INF = __builtin_inff();

typedef __attribute__((ext_vector_type(2))) float v2f;
typedef __attribute__((ext_vector_type(8))) float v8f;

// ---------------------------------------------------------------------------
// Kernel 0: initialize workspace (harness poisons d_ws with 0xAA)
// ---------------------------------------------------------------------------
__global__ void tl_init_kernel(unsigned* __restrict__ posb,
                               unsigned* __restrict__ negb,
                               float* __restrict__ accum) {
  int i = blockIdx.x * blockDim.x + threadIdx.x;
  if (i < N) {
    posb[i] = 0u;           // max over nonneg dists; 0.0f is identity
    negb[i] = 0x7F800000u;  // +inf bits; identity for min
  }
  if (i < 2) accum[i] = 0.0f;  // [0]=std_sum, [1]=xent_sum
}

// ---------------------------------------------------------------------------
// Kernel 1: per-row stats. One 256-thread block per row.
//   sq[row] = sum(x^2); accum[0] += std(row, ddof=1); accum[1] += -logp[target]
// ---------------------------------------------------------------------------
__global__ __launch_bounds__(256) void tl_row_stats_kernel(
    const float* __restrict__ X, const int* __restrict__ targets,
    float* __restrict__ sq, float* __restrict__ accum) {
  const int row = blockIdx.x;
  const int tid = threadIdx.x;
  const float* __restrict__ x = X + (size_t)row * D;

  float s = 0.f, ss = 0.f, mx = -F_INF;
  for (int c = tid; c < D; c += 256) {
    float v = x[c];
    s += v;
    ss += v * v;
    mx = fmaxf(mx, v);
  }

  __shared__ float sh0[256], sh1[256], sh2[256];
  sh0[tid] = s; sh1[tid] = ss; sh2[tid] = mx;
  __syncthreads();
  for (int w = 128; w > 0; w >>= 1) {
    if (tid < w) {
      sh0[tid] += sh0[tid + w];
      sh1[tid] += sh1[tid + w];
      sh2[tid] = fmaxf(sh2[tid], sh2[tid + w]);
    }
    __syncthreads();
  }
  const float stot = sh0[0], sstot = sh1[0], rmax = sh2[0];
  __syncthreads();  // everyone has rmax; safe to reuse sh0

  // second pass: sum(exp(x - rmax)) for log_softmax
  float se = 0.f;
  for (int c = tid; c < D; c += 256) se += expf(x[c] - rmax);
  sh0[tid] = se;
  __syncthreads();
  for (int w = 128; w > 0; w >>= 1) {
    if (tid < w) sh0[tid] += sh0[tid + w];
    __syncthreads();
  }

  if (tid == 0) {
    sq[row] = sstot;
    // unbiased variance: (sum(x^2) - sum(x)^2/D) / (D-1)
    float var = fmaxf(sstot - stot * (stot / (float)D), 0.f) / (float)(D - 1);
    atomicAdd(&accum[0], sqrtf(var));
    int t = targets[row];
    float logp_t = x[t] - rmax - logf(sh0[0]);
    atomicAdd(&accum[1], -logp_t);
  }
}

// ---------------------------------------------------------------------------
// Kernel 2: fused Gram GEMM (fp32 WMMA 16x16x4) + distance + hard mining.
// Block = 4 waves, 128x128 output tile; each wave owns a 64x64 subtile
// (4x4 grid of 16x16 WMMA accumulators). dist never hits memory.
// ---------------------------------------------------------------------------
__global__ __launch_bounds__(128) void tl_gram_mine_kernel(
    const float* __restrict__ X, const float* __restrict__ sq,
    unsigned* __restrict__ posb, unsigned* __restrict__ negb) {
  const int lane = threadIdx.x & 31;
  const int wave = threadIdx.x >> 5;
  const int wr = wave >> 1, wc = wave & 1;  // 2x2 waves in block
  const int i0 = (blockIdx.y * 2 + wr) * 64;
  const int j0 = (blockIdx.x * 2 + wc) * 64;
  const int half = lane >> 4;   // 0: lanes 0-15, 1: lanes 16-31
  const int lm   = lane & 15;
  const int ko   = half * 2;    // K sub-offset owned by this half-wave

  v8f acc[4][4];
#pragma unroll
  for (int r = 0; r < 4; ++r)
#pragma unroll
    for (int c = 0; c < 4; ++c) acc[r][c] = (v8f)0.0f;

  // Per f32 WMMA layout: A lane holds X[i0+r*16+lm][k+ko .. k+ko+1]
  //                      B lane holds X[j0+c*16+lm][k+ko .. k+ko+1]
  const float* arow[4];
  const float* brow[4];
#pragma unroll
  for (int r = 0; r < 4; ++r) arow[r] = X + (size_t)(i0 + r * 16 + lm) * D + ko;
#pragma unroll
  for (int c = 0; c < 4; ++c) brow[c] = X + (size_t)(j0 + c * 16 + lm) * D + ko;

  for (int k = 0; k < D; k += 4) {
    v2f a[4], b[4];
#pragma unroll
    for (int r = 0; r < 4; ++r) a[r] = *(const v2f*)(arow[r] + k);
#pragma unroll
    for (int c = 0; c < 4; ++c) b[c] = *(const v2f*)(brow[c] + k);
#pragma unroll
    for (int r = 0; r < 4; ++r)
#pragma unroll
      for (int c = 0; c < 4; ++c)
        acc[r][c] = __builtin_amdgcn_wmma_f32_16x16x4_f32(
            /*neg_a=*/false, a[r], /*neg_b=*/false, b[c],
            /*c_mod=*/(short)0, acc[r][c],
            /*reuse_a=*/false, /*reuse_b=*/false);
  }

  // Epilogue: dist = sqrt(clip(sq_i + sq_j - 2G, 1e-12)); mine per row.
  // C/D layout: VGPR v, lane L -> row m = v + 8*(L>>4), col n = L&15.
  float sqj[4];
#pragma unroll
  for (int c = 0; c < 4; ++c) sqj[c] = sq[j0 + c * 16 + lm];

#pragma unroll
  for (int r = 0; r < 4; ++r) {
#pragma unroll
    for (int v = 0; v < 8; ++v) {
      const int i = i0 + r * 16 + v + 8 * half;  // uniform across the 16 lanes
      const float sqi = sq[i];
      float pmax = 0.f, nmin = F_INF;
#pragma unroll
      for (int c = 0; c < 4; ++c) {
        const int j = j0 + c * 16 + lm;
        float g = acc[r][c][v];
        float d2 = sqi + sqj[c] - 2.0f * g;
        float dist = sqrtf(fmaxf(d2, 1e-12f));
        bool same = (i >> 2) == (j >> 2);  // targets = index/4
        pmax = fmaxf(pmax, same ? dist : 0.f);
        nmin = fminf(nmin, same ? F_INF : dist);
      }
      // reduce over the 16 lanes of this half (xor masks stay in-half)
#pragma unroll
      for (int off = 1; off < 16; off <<= 1) {
        pmax = fmaxf(pmax, __shfl_xor(pmax, off, 32));
        nmin = fminf(nmin, __shfl_xor(nmin, off, 32));
      }
      if (lm == 0) {
        // dist >= 0 so IEEE bit pattern is monotonic: integer atomics suffice
        atomicMax(&posb[i], __float_as_uint(pmax));
        atomicMin(&negb[i], __float_as_uint(nmin));
      }
    }
  }
}

// ---------------------------------------------------------------------------
// Kernel 3: finalize -> out[0]=loss, out[1]=precision
// ---------------------------------------------------------------------------
__global__ __launch_bounds__(256) void tl_finalize_kernel(
    const unsigned* __restrict__ posb, const unsigned* __restrict__ negb,
    const float* __restrict__ accum, float* __restrict__ out) {
  const int tid = threadIdx.x;
  float tsum = 0.f, pcnt = 0.f;
  for (int i = tid; i < N; i += 256) {
    float ap = __uint_as_float(posb[i]);
    float an = __uint_as_float(negb[i]);
    tsum += fmaxf(ap - an, 0.f);  // MARGIN = 0
    pcnt += (an > ap) ? 1.f : 0.f;
  }
  __shared__ float sh0[256], sh1[256];
  sh0[tid] = tsum; sh1[tid] = pcnt;
  __syncthreads();
  for (int w = 128; w > 0; w >>= 1) {
    if (tid < w) { sh0[tid] += sh0[tid + w]; sh1[tid] += sh1[tid + w]; }
    __syncthreads();
  }
  if (tid == 0) {
    const float invN = 1.0f / (float)N;
    float triplet = sh0[0] * invN;
    float loss = 1.0f * triplet + 0.5f * accum[0] + 0.5f * (accum[1] * invN);
    out[0] = loss;
    out[1] = sh1[0] * invN;
  }
}

// ---------------------------------------------------------------------------
extern "C" void kernel_launch(void* const* d_in, const int* in_sizes, int n_in,
                              void* d_out, int out_size, void* d_ws, size_t ws_size,
                              hipStream_t stream) {
  const float* X       = (const float*)d_in[0];  // [N, D] fp32
  const int*   targets = (const int*)d_in[1];    // [N] int32
  float* out = (float*)d_out;                    // {loss, prec}

  // workspace layout
  float*    sq    = (float*)d_ws;            // N floats
  unsigned* posb  = (unsigned*)(sq + N);     // N u32 (hardest-positive bits)
  unsigned* negb  = posb + N;                // N u32 (hardest-negative bits)
  float*    accum = (float*)(negb + N);      // [0]=std_sum, [1]=xent_sum

  tl_init_kernel<<<(N + 255) / 256, 256, 0, stream>>>(posb, negb, accum);
  tl_row_stats_kernel<<<N, 256, 0, stream>>>(X, targets, sq, accum);
  dim3 grid(N / 128, N / 128);               // 32 x 32 blocks of 128x128 tiles
  tl_gram_mine_kernel<<<grid, 128, 0, stream>>>(X, sq, posb, negb);
  tl_finalize_kernel<<<1, 256, 0, stream>>>(posb, negb, accum, out);
}